// FeatureAttentionLayer_83013127897470
// MI455X (gfx1250) — compile-verified
//
#include <hip/hip_runtime.h>
#include <hip/hip_bf16.h>
#include <math.h>

#define ALPHA 0.2f
#define Bb 16
#define Kk 128
#define Ww 256
#define Ee 512

typedef float v2f __attribute__((ext_vector_type(2)));
typedef float v8f __attribute__((ext_vector_type(8)));

// ---------------------------------------------------------------------------
// Kernel 1: Gumbel-thresholded cosine adjacency (K x K)
// ---------------------------------------------------------------------------
__global__ void adj_kernel(const float* __restrict__ emb,
                           const float* __restrict__ gumbel,
                           float* __restrict__ adj) {
    int idx = blockIdx.x * blockDim.x + threadIdx.x;
    if (idx >= Kk * Kk) return;
    int i = idx / Kk, j = idx % Kk;
    const float* ei = emb + (size_t)i * Ee;
    const float* ej = emb + (size_t)j * Ee;
    float dot = 0.f, ni = 0.f, nj = 0.f;
    for (int e = 0; e < Ee; ++e) {
        float va = ei[e], vb = ej[e];
        dot += va * vb; ni += va * va; nj += vb * vb;
    }
    float outer = sqrtf(ni) * sqrtf(nj);
    float lg = dot / outer / outer;
    lg = (lg + 1.0f) * 0.5f;
    float u0 = gumbel[(size_t)idx * 2 + 0];
    float u1 = gumbel[(size_t)idx * 2 + 1];
    float y0 = lg + (-logf(-logf(u0)));
    float y1 = (1.0f - lg) + (-logf(-logf(u1)));
    float v = (y0 > y1) ? 1.0f : 0.0f;
    if (i == j) v = 0.0f;
    adj[idx] = v;
}

// ---------------------------------------------------------------------------
// Kernel 2: L = v @ [W1 | W2] (+ b_lin on W2 half), fp32 WMMA 16x16x4.
// v[b][i][w] == x[b][w][i].  One wave per 16x16 output tile.
// ---------------------------------------------------------------------------
__global__ void __launch_bounds__(256) lin_kernel(const float* __restrict__ x,
                                                  const float* __restrict__ W_lin,
                                                  const float* __restrict__ b_lin,
                                                  float* __restrict__ L1,
                                                  float* __restrict__ L2) {
    int wave = threadIdx.x >> 5;
    int lane = threadIdx.x & 31;
    int tile = blockIdx.x * 8 + wave;          // 0..8191
    int b    = tile >> 9;                      // 16 batches
    int rem  = tile & 511;
    int mt   = rem >> 6;                       // 8 M-tiles (K=128 rows)
    int nt   = rem & 63;                       // 64 N-tiles (2*E cols)
    int i0   = mt * 16;
    int half = nt >> 5;                        // 0 -> W1, 1 -> W2
    int n0   = (nt & 31) * 16;                 // column within E
    int ncol = n0 + (lane & 15);
    int kh   = (lane >> 4) * 2;                // K-pair select within step-of-4
    int m    = lane & 15;

    const float* xb = x + (size_t)b * Ww * Kk;
    const float* Wm = W_lin + (half ? (size_t)Ww * Ee : 0);

    v8f c = {};
    for (int k = 0; k < Ww; k += 4) {
        int kk = k + kh;
        v2f av, bv;
        // A[m][k] = v[b][i0+m][k] = x[b][k][i0+m]
        av.x = xb[(size_t)kk       * Kk + i0 + m];
        av.y = xb[(size_t)(kk + 1) * Kk + i0 + m];
        // B[k][n] = W[k][n0+n]
        bv.x = Wm[(size_t)kk       * Ee + ncol];
        bv.y = Wm[(size_t)(kk + 1) * Ee + ncol];
        c = __builtin_amdgcn_wmma_f32_16x16x4_f32(false, av, false, bv,
                                                  (short)0, c, false, false);
    }
    float  biasn = half ? b_lin[ncol] : 0.0f;
    float* Ldst  = (half ? L2 : L1) + (size_t)b * Kk * Ee;
    int rbase = i0 + ((lane >> 4) << 3);       // C row = r + 8*(lane/16)
#pragma unroll
    for (int r = 0; r < 8; ++r)
        Ldst[(size_t)(rbase + r) * Ee + ncol] = c[r] + biasn;
}

// ---------------------------------------------------------------------------
// Kernel 3: fused  e -> mask -> softmax -> attn@v (WMMA) -> sigmoid -> store^T
// One block = (batch b, 16-row i-tile).  256 threads = 8 wave32.
// ---------------------------------------------------------------------------
__global__ void __launch_bounds__(256) attn_kernel(const float* __restrict__ x,
                                                   const float* __restrict__ adj,
                                                   const float* __restrict__ bias,
                                                   const float* __restrict__ a,
                                                   const float* __restrict__ L1,
                                                   const float* __restrict__ L2,
                                                   float* __restrict__ out) {
    __shared__ float sL1[16 * Ee];   // 32 KB: L1 tile rows i0..i0+15
    __shared__ float sA[Ee];         //  2 KB: vector a
    __shared__ float eS[16 * Kk];    //  8 KB: logits -> attn

    int t  = threadIdx.x;
    int b  = blockIdx.x >> 3;
    int it = blockIdx.x & 7;
    int i0 = it * 16;

    const float* L1b = L1 + ((size_t)b * Kk + i0) * Ee;  // 16 contiguous rows
    const float* L2b = L2 + (size_t)b * Kk * Ee;

    for (int p = t; p < 16 * Ee; p += 256) sL1[p] = L1b[p];
    for (int p = t; p < Ee;      p += 256) sA[p]  = a[p];
    __syncthreads();

    // ---- e[m][j] = sum_e a_e * leaky(L1[i0+m][e] + L2[j][e]) --------------
    int m  = t & 15;      // row within tile
    int jg = t >> 4;      // 16 groups of 8 columns
    float acc[8] = {0, 0, 0, 0, 0, 0, 0, 0};
    for (int e = 0; e < Ee; ++e) {
        float l1 = sL1[m * Ee + e];
        float av = sA[e];
#pragma unroll
        for (int q = 0; q < 8; ++q) {
            float pre = l1 + L2b[(size_t)(jg * 8 + q) * Ee + e];
            float act = pre > 0.f ? pre : ALPHA * pre;
            acc[q] += av * act;
        }
    }
#pragma unroll
    for (int q = 0; q < 8; ++q) {
        int j = jg * 8 + q;
        float ad = adj[(size_t)(i0 + m) * Kk + j];
        float ev = (ad == 1.0f) ? acc[q] : -1.0e12f;
        eS[m * Kk + j] = ev + bias[(size_t)(i0 + m) * Kk + j];
    }
    __syncthreads();

    // ---- row softmax over j (threads 0..15, one row each) -----------------
    if (t < 16) {
        float mx = -3.0e38f;
        for (int j = 0; j < Kk; ++j) mx = fmaxf(mx, eS[t * Kk + j]);
        float s = 0.f;
        for (int j = 0; j < Kk; ++j) {
            float ex = expf(eS[t * Kk + j] - mx);
            eS[t * Kk + j] = ex;
            s += ex;
        }
        float inv = 1.0f / s;
        for (int j = 0; j < Kk; ++j) eS[t * Kk + j] *= inv;
    }
    __syncthreads();

    // ---- h = sigmoid(attn @ v), store transposed: out[b][w][i] ------------
    int wave = t >> 5, lane = t & 31;
    int kh = (lane >> 4) * 2;
    int ln = lane & 15;                 // A row m / B col n
    const float* xb = x + (size_t)b * Ww * Kk;
    for (int s = 0; s < 2; ++s) {
        int w0 = (wave * 2 + s) * 16;   // 16 w-tiles, 2 per wave
        v8f c = {};
        for (int k = 0; k < Kk; k += 4) {
            v2f av, bv;
            // A[m][k] = attn[m][k]  (from LDS)
            av.x = eS[ln * Kk + k + kh];
            av.y = eS[ln * Kk + k + kh + 1];
            // B[k][n] = v[b][k][w0+n] = x[b][w0+n][k]  (contiguous pair)
            bv.x = xb[(size_t)(w0 + ln) * Kk + k + kh];
            bv.y = xb[(size_t)(w0 + ln) * Kk + k + kh + 1];
            c = __builtin_amdgcn_wmma_f32_16x16x4_f32(false, av, false, bv,
                                                      (short)0, c, false, false);
        }
        int rbase = (lane >> 4) << 3;
#pragma unroll
        for (int r = 0; r < 8; ++r) {
            float h = 1.0f / (1.0f + expf(-c[r]));
            out[(size_t)b * Ww * Kk + (size_t)(w0 + ln) * Kk + i0 + rbase + r] = h;
        }
    }
}

// ---------------------------------------------------------------------------
extern "C" void kernel_launch(void* const* d_in, const int* in_sizes, int n_in,
                              void* d_out, int out_size, void* d_ws, size_t ws_size,
                              hipStream_t stream) {
    const float* x      = (const float*)d_in[0];  // (B, W, K)
    const float* emb    = (const float*)d_in[1];  // (K, E)
    const float* W_lin  = (const float*)d_in[2];  // (2W, E)
    const float* b_lin  = (const float*)d_in[3];  // (E,)
    const float* a      = (const float*)d_in[4];  // (E,)
    const float* bias   = (const float*)d_in[5];  // (K, K)
    const float* gumbel = (const float*)d_in[6];  // (K, K, 2)
    float* out = (float*)d_out;                   // (B, W, K)

    float* adj = (float*)d_ws;                    // K*K
    float* L1  = adj + (size_t)Kk * Kk;           // B*K*E
    float* L2  = L1 + (size_t)Bb * Kk * Ee;       // B*K*E

    adj_kernel<<<(Kk * Kk + 255) / 256, 256, 0, stream>>>(emb, gumbel, adj);
    lin_kernel<<<1024, 256, 0, stream>>>(x, W_lin, b_lin, L1, L2);
    attn_kernel<<<Bb * 8, 256, 0, stream>>>(x, adj, bias, a, L1, L2, out);
}